// KVCache_17789754540260
// MI455X (gfx1250) — compile-verified
//
#include <hip/hip_runtime.h>

// ---------------------------------------------------------------------------
// KV cache append (concat along seq axis). Pure streaming copy:
//   out_k[b,h,0:S,:]    = k_cache ; out_k[b,h,S:S+Q,:] = k_val   (same for V)
// ~2.15 GB of traffic, 0 FLOPs -> HBM-bound (~92us at 23.3 TB/s).
// Bulk path uses gfx1250 async global<->LDS DMA (ASYNCcnt) when the builtins
// exist; otherwise non-temporal B128 load/store. Both are full-bandwidth.
// ---------------------------------------------------------------------------

typedef float f4  __attribute__((ext_vector_type(4)));
typedef int   v4i __attribute__((ext_vector_type(4)));

#define AS1 __attribute__((address_space(1)))
#define AS3 __attribute__((address_space(3)))

// Problem constants (from reference): B=4, H=32, S=8192, Q=16, D=128
constexpr size_t BH    = 4ull * 32;                 // 128 (b,h) blocks
constexpr size_t SD4   = 8192ull * 128 / 4;         // 262144 = 2^18 float4 per cache block
constexpr size_t QD4   = 16ull * 128 / 4;           // 512 float4 per val block
constexpr size_t OUT4  = (8192ull + 16) * 128 / 4;  // 262656 float4 per out block
constexpr size_t TOTC4 = BH * SD4;                  // 2^25 cache float4 per tensor
constexpr size_t TOTV4 = BH * QD4;                  // 2^16 val float4 per tensor
constexpr size_t TOTO4 = BH * OUT4;                 // out float4 per tensor

#if defined(__HIP_DEVICE_COMPILE__) &&                                   \
    __has_builtin(__builtin_amdgcn_global_load_async_to_lds_b128) &&     \
    __has_builtin(__builtin_amdgcn_global_store_async_from_lds_b128)
#define USE_ASYNC_LDS 1
#else
#define USE_ASYNC_LDS 0
#endif

__device__ __forceinline__ void wait_async0() {
#if defined(__HIP_DEVICE_COMPILE__)
#if __has_builtin(__builtin_amdgcn_s_wait_asynccnt)
  __builtin_amdgcn_s_wait_asynccnt(0);
#else
  asm volatile("s_wait_asynccnt 0" ::: "memory");
#endif
#endif
}

// ---------------- bulk: cache portion of K and V (2^26 float4) -------------
__global__ __launch_bounds__(256) void kv_bulk_copy(const f4* __restrict__ kc,
                                                    const f4* __restrict__ vc,
                                                    f4* __restrict__ out) {
#if USE_ASYNC_LDS
  // 8 waves x 256 float4 (4KB) per-wave staging = 32KB LDS per workgroup.
  __shared__ f4 stage[8 * 256];
  const unsigned wave = threadIdx.x >> 5;
  const unsigned lane = threadIdx.x & 31u;
  f4* wbuf = &stage[wave * 256];
  const size_t wavesTotal = (size_t)gridDim.x * (blockDim.x >> 5);
  const size_t waveId     = (size_t)blockIdx.x * (blockDim.x >> 5) + wave;
  constexpr size_t TILE = 256;  // float4 per wave tile (8 chunks x 32 lanes)
  for (size_t base = waveId * TILE; base < 2 * TOTC4; base += wavesTotal * TILE) {
#pragma unroll
    for (int u = 0; u < 8; ++u) {
      size_t i = base + (size_t)u * 32 + lane;
      size_t t = i >> 25;            // 0 = K, 1 = V   (TOTC4 == 2^25)
      size_t j = i & (TOTC4 - 1);
      const f4* src = t ? (vc + j) : (kc + j);
      __builtin_amdgcn_global_load_async_to_lds_b128(
          (AS1 v4i*)src, (AS3 v4i*)(wbuf + u * 32 + lane), 0, 0);
    }
    wait_async0();  // LDS staging complete
#pragma unroll
    for (int u = 0; u < 8; ++u) {
      size_t i   = base + (size_t)u * 32 + lane;
      size_t t   = i >> 25;
      size_t j   = i & (TOTC4 - 1);
      size_t blk = j >> 18;          // SD4 == 2^18
      size_t r   = j & (SD4 - 1);
      f4* dst = out + t * TOTO4 + blk * OUT4 + r;
      __builtin_amdgcn_global_store_async_from_lds_b128(
          (AS1 v4i*)dst, (AS3 v4i*)(wbuf + u * 32 + lane), 0, 0);
    }
    wait_async0();  // stores drained before LDS reuse
  }
#else
  const size_t stride = (size_t)gridDim.x * blockDim.x;
  for (size_t i = (size_t)blockIdx.x * blockDim.x + threadIdx.x; i < 2 * TOTC4;
       i += stride) {
    size_t t   = i >> 25;            // 0 = K, 1 = V
    size_t j   = i & (TOTC4 - 1);
    size_t blk = j >> 18;
    size_t r   = j & (SD4 - 1);
    const f4* src = t ? vc : kc;
    f4 v = __builtin_nontemporal_load(src + j);           // streamed once
    __builtin_nontemporal_store(v, out + t * TOTO4 + blk * OUT4 + r);
  }
#endif
}

// ---------------- tail: appended K/V rows (2 * 2^16 float4, ~2 MB) ---------
__global__ __launch_bounds__(256) void kv_val_copy(const f4* __restrict__ kv,
                                                   const f4* __restrict__ vv,
                                                   f4* __restrict__ out) {
  size_t i = (size_t)blockIdx.x * blockDim.x + threadIdx.x;
  if (i >= 2 * TOTV4) return;
  size_t t   = i >> 16;              // TOTV4 == 2^16
  size_t j   = i & (TOTV4 - 1);
  size_t blk = j >> 9;               // QD4 == 2^9
  size_t r   = j & (QD4 - 1);
  const f4* src = t ? vv : kv;
  out[t * TOTO4 + blk * OUT4 + SD4 + r] = src[j];
}

extern "C" void kernel_launch(void* const* d_in, const int* in_sizes, int n_in,
                              void* d_out, int out_size, void* d_ws,
                              size_t ws_size, hipStream_t stream) {
  (void)in_sizes; (void)n_in; (void)out_size; (void)d_ws; (void)ws_size;
  const f4* kc = (const f4*)d_in[0];
  const f4* vc = (const f4*)d_in[1];
  const f4* kv = (const f4*)d_in[2];
  const f4* vv = (const f4*)d_in[3];
  f4* out = (f4*)d_out;

  // Bulk: 2^26 float4. 8192 blocks x 256 threads; per-wave tiles (async path)
  // or grid-stride (fallback) cover the range exactly.
  kv_bulk_copy<<<8192, 256, 0, stream>>>(kc, vc, out);

  // Tail: 131072 float4 -> exact-sized launch.
  kv_val_copy<<<512, 256, 0, stream>>>(kv, vv, out);
}